// GraphAttentionLayer_67010079752626
// MI455X (gfx1250) — compile-verified
//
#include <hip/hip_runtime.h>

typedef __attribute__((ext_vector_type(2))) float v2f;
typedef __attribute__((ext_vector_type(4))) float v4f;
typedef __attribute__((ext_vector_type(8))) float v8f;

#define N_NODES 20000
#define DEG     16
#define IN_F    256
#define HEADS   8
#define D_HEAD  64
#define OUT_F   (HEADS * D_HEAD)   // 512
#define M_TILES (N_NODES / 16)     // 1250

// ---------------------------------------------------------------------------
// Kernel 1: h = x @ W   (20000x256 @ 256x512, f32, WMMA 16x16x4)
// Block = 8 waves = 128(M) x 64(N) super-tile; W strip staged in LDS once and
// reused by all 8 waves (8x cut in L2 B-traffic). Each wave: 16x64 strip with
// 4 independent WMMA accumulator chains; A from global, B via ds_load_b64.
//
// LDS layout (K-group-of-4 swizzle, conflict-free B fragments):
//   wlds[(k>>2)*256 + c*4 + (k&3)]   (c = column within the 64-col strip)
// B frag = contiguous {k=2*klane, 2*klane+1} pair at g*256 + c*4 + 2*klane:
//   lanes 0-15 banks {4c,4c+1}, lanes 16-31 banks {4c+2,4c+3} -> 64 banks.
//
// Fragment layouts (ISA 7.12.2, 32-bit):
//   A 16x4 : lanes 0-15 row M=lane (K=k,k+1), lanes 16-31 (K=k+2,k+3)
//   B 4x16 : lanes 0-15 col N=lane (K=k,k+1), lanes 16-31 (K=k+2,k+3)
//   C/D    : VGPR r -> M = r + 8*(lane>=16), N = lane&15
// ---------------------------------------------------------------------------
__global__ __launch_bounds__(256) void gat_gemm_h(const float* __restrict__ x,
                                                  const float* __restrict__ W,
                                                  float* __restrict__ h)
{
    __shared__ float wlds[IN_F * 64];                // 64 KB

    const int t    = threadIdx.x;
    const int wave = t >> 5;
    const int lane = t & 31;
    const int ts   = blockIdx.x & 7;                 // N strip 0..7 (cols ts*64..)
    const int tmg  = blockIdx.x >> 3;                // M group 0..156
    const int tm   = tmg * 8 + wave;                 // wave's M tile (may be OOB)

    // ---- stage W[:, ts*64 : ts*64+64] into LDS (all threads) ----
    #pragma unroll
    for (int i = 0; i < 16; ++i) {
        const int idx4 = i * 256 + t;                // 0..4095 float4s
        const int k    = idx4 >> 4;                  // K row 0..255
        const int c4   = idx4 & 15;                  // float4 within row
        const v4f v = *(const v4f*)(W + (long)k * OUT_F + ts * 64 + c4 * 4);
        float* __restrict__ dst = wlds + (k >> 2) * 256 + (k & 3);
        dst[(c4 * 4 + 0) * 4] = v.x;
        dst[(c4 * 4 + 1) * 4] = v.y;
        dst[(c4 * 4 + 2) * 4] = v.z;
        dst[(c4 * 4 + 3) * 4] = v.w;
    }
    __syncthreads();

    if (tm < M_TILES) {                              // wave-uniform guard
        const int klane = lane >> 4;                 // 0 or 1
        const int mlane = lane & 15;
        const float* __restrict__ xr = x + (long)(tm * 16 + mlane) * IN_F;

        v8f acc0 = {}, acc1 = {}, acc2 = {}, acc3 = {};

        #pragma unroll 2
        for (int k = 0; k < IN_F; k += 4) {
            const int ka = k + 2 * klane;
            v2f a;
            a.x = xr[ka];
            a.y = xr[ka + 1];

            const float* __restrict__ bl =
                wlds + (k >> 2) * 256 + mlane * 4 + 2 * klane;

            v2f b;
            b = *(const v2f*)(bl + 0);               // cols  0..15
            acc0 = __builtin_amdgcn_wmma_f32_16x16x4_f32(false, a, false, b,
                                                         (short)0, acc0, false, false);
            b = *(const v2f*)(bl + 64);              // cols 16..31
            acc1 = __builtin_amdgcn_wmma_f32_16x16x4_f32(false, a, false, b,
                                                         (short)0, acc1, false, false);
            b = *(const v2f*)(bl + 128);             // cols 32..47
            acc2 = __builtin_amdgcn_wmma_f32_16x16x4_f32(false, a, false, b,
                                                         (short)0, acc2, false, false);
            b = *(const v2f*)(bl + 192);             // cols 48..63
            acc3 = __builtin_amdgcn_wmma_f32_16x16x4_f32(false, a, false, b,
                                                         (short)0, acc3, false, false);
        }

        float* __restrict__ hc =
            h + (long)(tm * 16 + 8 * klane) * OUT_F + ts * 64 + mlane;
        #pragma unroll
        for (int r = 0; r < 8; ++r) {
            float* __restrict__ hr = hc + (long)r * OUT_F;
            hr[0]  = acc0[r];
            hr[16] = acc1[r];
            hr[32] = acc2[r];
            hr[48] = acc3[r];
        }
    }
}

// ---------------------------------------------------------------------------
// Kernel 2: attention logits, one wave per (node, head).
// Lanes load hv[lane] / hv[lane+32] (coalesced), FMA against 4KB-cached `a`,
// wave32 shuffle-tree reduction, lane 0 writes.
// ---------------------------------------------------------------------------
__global__ __launch_bounds__(256) void gat_alpha(const float* __restrict__ h,
                                                 const float* __restrict__ a,
                                                 float* __restrict__ alpha_src,
                                                 float* __restrict__ alpha_dst)
{
    const int wave = threadIdx.x >> 5;
    const int lane = threadIdx.x & 31;
    const long tpair = (long)blockIdx.x * 8 + wave;   // (node, head) pair id
    if (tpair >= (long)N_NODES * HEADS) return;       // uniform per wave
    const int  head = (int)(tpair % HEADS);
    const long n    = tpair / HEADS;

    const float* __restrict__ hv = h + n * OUT_F + head * D_HEAD;
    const float* __restrict__ as = a + head * 2 * D_HEAD;
    const float* __restrict__ ad = as + D_HEAD;

    const float v0 = hv[lane];
    const float v1 = hv[lane + 32];
    float s = fmaf(v0, as[lane], v1 * as[lane + 32]);
    float d = fmaf(v0, ad[lane], v1 * ad[lane + 32]);

    #pragma unroll
    for (int off = 16; off; off >>= 1) {
        s += __shfl_xor(s, off, 32);
        d += __shfl_xor(d, off, 32);
    }
    if (lane == 0) {
        alpha_src[tpair] = s;
        alpha_dst[tpair] = d;
    }
}

// ---------------------------------------------------------------------------
// Kernel 3: softmax over 16 neighbors + weighted gather + ReLU.
// One block per node; wave w handles head w; each lane owns 2 feature dims.
// h (41 MB) is L2-resident (192 MB L2) so the 655 MB gather stays on-chip.
// Reference semantics: m = max(max_j e_j, 0),
//   denom = sum_j exp(e_j - m) + (N - deg) * exp(-m),  deg == 16 for all nodes.
// ---------------------------------------------------------------------------
__global__ __launch_bounds__(256) void gat_aggregate(const float* __restrict__ h,
                                                     const float* __restrict__ alpha_src,
                                                     const float* __restrict__ alpha_dst,
                                                     const int*   __restrict__ col,
                                                     float* __restrict__ out)
{
    __shared__ int scol[DEG];
    const int n    = blockIdx.x;
    const int head = threadIdx.x >> 5;
    const int lane = threadIdx.x & 31;

    if (threadIdx.x < DEG)
        scol[threadIdx.x] = col[(long)n * DEG + threadIdx.x];
    __syncthreads();

    const float asrc_n = alpha_src[n * HEADS + head];

    float e = -__builtin_inff();                      // lanes >= DEG stay -inf
    if (lane < DEG)
        e = asrc_n + alpha_dst[scol[lane] * HEADS + head];

    float m = e;
    #pragma unroll
    for (int off = 16; off; off >>= 1)
        m = fmaxf(m, __shfl_xor(m, off, 32));
    m = fmaxf(m, 0.0f);

    const float p = (lane < DEG) ? __expf(e - m) : 0.0f;
    float s = p;
    #pragma unroll
    for (int off = 16; off; off >>= 1)
        s += __shfl_xor(s, off, 32);

    const float denom = s + (float)(N_NODES - DEG) * __expf(-m);
    const float inv   = 1.0f / denom;

    float acc0 = 0.0f, acc1 = 0.0f;
    #pragma unroll
    for (int j = 0; j < DEG; ++j) {
        const float attn = __shfl(p, j, 32) * inv;    // broadcast exp(e_j - m)
        const float* __restrict__ hv =
            h + (long)scol[j] * OUT_F + head * D_HEAD;
        acc0 = fmaf(attn, hv[lane],      acc0);
        acc1 = fmaf(attn, hv[lane + 32], acc1);
    }

    float* __restrict__ o = out + (long)n * OUT_F + head * D_HEAD;
    o[lane]      = fmaxf(acc0, 0.0f);
    o[lane + 32] = fmaxf(acc1, 0.0f);
}

// ---------------------------------------------------------------------------
extern "C" void kernel_launch(void* const* d_in, const int* in_sizes, int n_in,
                              void* d_out, int out_size, void* d_ws, size_t ws_size,
                              hipStream_t stream)
{
    const float* x    = (const float*)d_in[0];                 // (20000, 256)
    const float* W    = (const float*)d_in[1];                 // (256, 512)
    const float* a    = (const float*)d_in[2];                 // (1, 8, 128)
    const int*   edge = (const int*)d_in[3];                   // (2, 320000)
    float*       out  = (float*)d_out;                         // (20000, 512)

    // workspace: h (40.96 MB) | alpha_src (640 KB) | alpha_dst (640 KB)
    float* h         = (float*)d_ws;
    float* alpha_src = h + (size_t)N_NODES * OUT_F;
    float* alpha_dst = alpha_src + (size_t)N_NODES * HEADS;

    const int* col = edge + (size_t)N_NODES * DEG;   // edge_index row 1

    // 157 M-groups (ceil(1250/8)) x 8 N-strips
    const int mgroups = (M_TILES + 7) / 8;           // 157
    gat_gemm_h<<<mgroups * 8, 256, 0, stream>>>(x, W, h);

    // one wave per (node, head): 160000 waves, 8 waves/block
    gat_alpha<<<(int)(((long)N_NODES * HEADS + 7) / 8), 256, 0, stream>>>(
        h, a, alpha_src, alpha_dst);

    gat_aggregate<<<N_NODES, 256, 0, stream>>>(h, alpha_src, alpha_dst, col, out);
}